// SliceRankingLoss_74723841015851
// MI455X (gfx1250) — compile-verified
//
#include <hip/hip_runtime.h>
#include <hip/hip_bf16.h>

// ---------------------------------------------------------------------------
// SliceRankingLoss on MI455X (gfx1250, wave32).
// D[i,j] = softplus(l[j]-l[i]) is never materialized: each wave computes a
// 16x4 f32 tile of D per step and feeds V_WMMA_F32_16X16X4_F32 against a
// 4x16 binary mask matrix (cols 0..8 = group&nontoxic, col 9 = nontoxic).
// Accumulator C = y[i, n] = sum_j D[i,j] * V[j,n]; left-side toxic/group
// dots then give the 27 bilinear sums needed by the loss.
// Softplus is computed in native base-2 (v_exp_f32 / v_log_f32 directly,
// no ln-correction fixup code); the column mask (toxic -> 0) is precomputed
// outside the hot loop so the B operand costs only shr+and+cvt per element.
// ---------------------------------------------------------------------------

#define BNUM 8192
#define NGRP 9
#define TOXBIT 0x200u
#define TILE_COLS 4096
#define MAIN_BLOCKS 64   // 64 blocks * 8 waves * 16 rows = 8192 rows

typedef float v2f __attribute__((ext_vector_type(2)));
typedef float v8f __attribute__((ext_vector_type(8)));

#define LOG2E 1.4426950408889634f
#define LN2   0.6931471805599453f

__device__ __forceinline__ float softplusf(float x) {
    // softplus(x) = max(x,0) + ln2 * log2(1 + 2^(-|x|*log2e))
    // exp2 arg <= 0, log2 arg in [1,2] -> raw HW ops are safe & accurate.
    float e = __builtin_amdgcn_exp2f(-fabsf(x) * LOG2E);
    return fmaxf(x, 0.0f) + LN2 * __builtin_amdgcn_logf(1.0f + e);
}

// --- kernel 1: per-row flags (bits 0..8 group, bit 9 toxic) and the
//     precomputed B-operand column mask (0 for toxic cols, else groups|z-bit).
__global__ __launch_bounds__(256) void srl_flags_kernel(
    const float* __restrict__ y_tox, const float* __restrict__ y_id,
    unsigned* __restrict__ flags, unsigned* __restrict__ colmask) {
    int i = blockIdx.x * blockDim.x + threadIdx.x;
    if (i < BNUM) {
        unsigned f = (y_tox[i] >= 0.5f) ? TOXBIT : 0u;
#pragma unroll
        for (int g = 0; g < NGRP; ++g)
            f |= (y_id[i * NGRP + g] >= 0.5f) ? (1u << g) : 0u;
        flags[i] = f;
        colmask[i] = (f & TOXBIT) ? 0u : (f | TOXBIT);
    }
}

// --- kernel 2: WMMA sweep. Each wave owns a 16-row strip of D.
__global__ __launch_bounds__(256) void srl_main_kernel(
    const float* __restrict__ logits, const unsigned* __restrict__ flags,
    const unsigned* __restrict__ colmask, float* __restrict__ partials) {
    __shared__ uint2 sCol[TILE_COLS];   // {logit bits, colmask} per column
    __shared__ float sPart[8][27];

    const int tid  = threadIdx.x;
    const int wave = tid >> 5;
    const int lane = tid & 31;
    const int half = lane >> 4;   // K-half for A/B operands
    const int n    = lane & 15;   // A row (M) and C column (N)

    const int strip   = blockIdx.x * 8 + wave;
    const int rowBase = strip * 16;
    const float lrow  = logits[rowBase + n];   // A-operand row logit (M = lane&15)

    v8f acc = {};   // C: y[rowBase + vgpr + 8*half, n]

    for (int p = 0; p < 2; ++p) {
        const int base = p * TILE_COLS;
        __syncthreads();
        for (int j = tid; j < TILE_COLS; j += 256) {
            uint2 v;
            v.x = __float_as_uint(logits[base + j]);
            v.y = colmask[base + j];
            sCol[j] = v;
        }
        __syncthreads();

#pragma unroll 8
        for (int j0 = 0; j0 < TILE_COLS; j0 += 4) {
            const int j = j0 + 2 * half;          // K index for this lane
            const uint2 c0 = sCol[j];
            const uint2 c1 = sCol[j + 1];
            // A tile: D[rowBase + n, base + j + {0,1}]
            float a0 = softplusf(__uint_as_float(c0.x) - lrow);
            float a1 = softplusf(__uint_as_float(c1.x) - lrow);
            // B tile: precomputed column mask, bit n
            float b0 = (float)((c0.y >> n) & 1u);
            float b1 = (float)((c1.y >> n) & 1u);
            v2f A  = {a0, a1};
            v2f Bv = {b0, b1};
            acc = __builtin_amdgcn_wmma_f32_16x16x4_f32(
                false, A, false, Bv, (short)0, acc, false, false);
        }
    }

    // Left-side dots: S11[g] = u_g'.y[:,g], r[g] = w'.y[:,g], c[g] = u_g'.y[:,9]
    float m1_l = 0.0f, rr_l = 0.0f;
    float c_l[NGRP] = {0, 0, 0, 0, 0, 0, 0, 0, 0};
#pragma unroll
    for (int r = 0; r < 8; ++r) {
        const int i = rowBase + r + 8 * half;   // C layout: M = vgpr + 8*half
        const unsigned f = flags[i];
        const float y = acc[r];
        const bool tox = (f & TOXBIT) != 0u;
        const float yt = tox ? y : 0.0f;
        rr_l += yt;                               // w' D v_n      (n<9 used)
        m1_l += ((f >> n) & 1u) ? yt : 0.0f;      // u_n' D v_n    (n<9 used)
        if (n == 9 && tox) {                      // u_g' D z from column 9
#pragma unroll
            for (int g = 0; g < NGRP; ++g)
                if ((f >> g) & 1u) c_l[g] += y;
        }
    }
    // combine the two half-wave row sets (M 0..7 and 8..15)
    m1_l += __shfl_xor(m1_l, 16, 32);
    rr_l += __shfl_xor(rr_l, 16, 32);
#pragma unroll
    for (int g = 0; g < NGRP; ++g) c_l[g] += __shfl_xor(c_l[g], 16, 32);

    if (half == 0) {
        if (n < NGRP) {
            sPart[wave][n] = m1_l;
            sPart[wave][NGRP + n] = rr_l;
        } else if (n == 9) {
#pragma unroll
            for (int g = 0; g < NGRP; ++g) sPart[wave][18 + g] = c_l[g];
        }
    }
    __syncthreads();
    if (tid < 27) {
        float s = 0.0f;
        for (int w = 0; w < 8; ++w) s += sPart[w][tid];   // fixed order
        partials[blockIdx.x * 27 + tid] = s;
    }
}

// --- kernel 3: deterministic reduction over blocks + scalar loss epilogue
__global__ __launch_bounds__(256) void srl_final_kernel(
    const unsigned* __restrict__ flags, const float* __restrict__ partials,
    float* __restrict__ out, int nBlocks) {
    __shared__ int sSP[NGRP];   // |tox & group_g|
    __shared__ int sSN[NGRP];   // |~tox & group_g|
    __shared__ int sTox;
    __shared__ float S[27];

    const int tid = threadIdx.x;
    if (tid < NGRP) { sSP[tid] = 0; sSN[tid] = 0; }
    if (tid == 0) sTox = 0;
    __syncthreads();

    int lsp[NGRP] = {0}, lsn[NGRP] = {0}, ltox = 0;
    for (int i = tid; i < BNUM; i += 256) {
        const unsigned f = flags[i];
        const bool tox = (f & TOXBIT) != 0u;
        ltox += tox ? 1 : 0;
#pragma unroll
        for (int g = 0; g < NGRP; ++g) {
            const int a = (int)((f >> g) & 1u);
            if (tox) lsp[g] += a; else lsn[g] += a;
        }
    }
#pragma unroll
    for (int g = 0; g < NGRP; ++g) {
        atomicAdd(&sSP[g], lsp[g]);   // integer: order-independent
        atomicAdd(&sSN[g], lsn[g]);
    }
    atomicAdd(&sTox, ltox);

    if (tid < 27) {
        float s = 0.0f;
        for (int b = 0; b < nBlocks; ++b) s += partials[b * 27 + tid];
        S[tid] = s;
    }
    __syncthreads();

    if (tid == 0) {
        const int totTox = sTox;
        const int totNon = BNUM - totTox;
        float sum_gl4 = 0.0f;
        int n_gv = 0;
        for (int g = 0; g < NGRP; ++g) {
            const float s11 = S[g];
            const float rr  = S[NGRP + g];
            const float cc  = S[18 + g];
            const float num[3] = {s11, rr - s11, cc - s11};
            const int sp = sSP[g], sn = sSN[g];
            const int bp = totTox - sp, bn = totNon - sn;
            const float cnt[3] = {(float)sp * (float)sn,
                                  (float)bp * (float)sn,
                                  (float)sp * (float)bn};
            float gl = 0.0f;
            int nv = 0;
            for (int t = 0; t < 3; ++t) {
                if (cnt[t] > 0.0f) {
                    gl += num[t] / fmaxf(cnt[t], 1.0f);
                    nv++;
                }
            }
            gl /= fmaxf((float)nv, 1.0f);
            if (nv > 0) {
                const float g2 = gl * gl;
                sum_gl4 += g2 * g2;   // POWER_P = 4
                n_gv++;
            }
        }
        const float mean_p = sum_gl4 / fmaxf((float)n_gv, 1.0f);
        const float loss = sqrtf(sqrtf(mean_p));   // mean_p^(1/4)
        out[0] = (n_gv > 0) ? loss : 0.0f;
    }
}

extern "C" void kernel_launch(void* const* d_in, const int* in_sizes, int n_in,
                              void* d_out, int out_size, void* d_ws, size_t ws_size,
                              hipStream_t stream) {
    const float* logits = (const float*)d_in[0];   // (B,1) f32
    const float* y_tox  = (const float*)d_in[1];   // (B,1) f32
    const float* y_id   = (const float*)d_in[2];   // (B,9) f32

    unsigned* flags    = (unsigned*)d_ws;                                        // 32 KB
    unsigned* colmask  = (unsigned*)((char*)d_ws + BNUM * sizeof(unsigned));     // 32 KB
    float*    partials = (float*)((char*)d_ws + 2 * BNUM * sizeof(unsigned));    // 64*27 f32

    srl_flags_kernel<<<BNUM / 256, 256, 0, stream>>>(y_tox, y_id, flags, colmask);
    srl_main_kernel<<<MAIN_BLOCKS, 256, 0, stream>>>(logits, flags, colmask,
                                                     partials);
    srl_final_kernel<<<1, 256, 0, stream>>>(flags, partials, (float*)d_out,
                                            MAIN_BLOCKS);
}